// PhysicsEnergyLoss_39135742001384
// MI455X (gfx1250) — compile-verified
//
#include <hip/hip_runtime.h>
#include <hip/hip_bf16.h>

// ---------------------------------------------------------------------------
// PhysicsEnergyLoss on gfx1250 (MI455X, wave32, WMMA).
//
// collision term: 16x16 tiles of pairwise dot products via
//   V_WMMA_F32_16X16X4_F32  (A = 16x4 tile of pos_i [x,y,z,0],
//                            B = 4x16 tile of pos_j), then
//   ||pi-pj||^2 = ni + nj - 2*dot  elementwise on the 8-VGPR accumulator.
// log(a/b) computed as (log2 a - log2 b)*ln2 on the TRANS pipe (no IEEE div).
// ---------------------------------------------------------------------------

typedef __attribute__((ext_vector_type(2))) float v2f;
typedef __attribute__((ext_vector_type(8))) float v8f;

#define GRAVITY_W   0.1f
#define GROUND_W    0.1f
#define COLLISION_W 0.3f
#define ALPHA       0.1f
#define EPSILON     1e-5f
#define CONSTANT    0.01f
#define LN2F        0.6931471805599453f

#define BATCH   4
#define NPTS    4096
#define TILES   (NPTS / 16)   // 256 row/col tiles
#define NWAVES  8
#define TPB     (NWAVES * 32) // 256 threads

// --- gravity + ground terms: O(B*N) reduction --------------------------------
__global__ __launch_bounds__(256)
void grav_ground_kernel(const float* __restrict__ pos,
                        const float* __restrict__ rad,
                        float* __restrict__ out) {
    int idx = blockIdx.x * blockDim.x + threadIdx.x;   // [0, B*N)
    float y = pos[idx * 3 + 1];
    float r = rad[idx];
    float relu = fmaxf(r - y, 0.0f);                   // Y_GROUND == 0
    float v = (GRAVITY_W * y + GROUND_W * relu) * (1.0f / (float)(BATCH * NPTS));

    __shared__ float red[256];
    red[threadIdx.x] = v;
    __syncthreads();
    #pragma unroll
    for (int s = 128; s > 0; s >>= 1) {
        if (threadIdx.x < s) red[threadIdx.x] += red[threadIdx.x + s];
        __syncthreads();
    }
    if (threadIdx.x == 0) atomicAdd(out, red[0]);
}

// --- one 16x16 tile: WMMA dot products + barrier epilogue --------------------
template <bool DIAG>
__device__ __forceinline__
float tile_eval(const float* __restrict__ pb, const float* __restrict__ rb,
                int i0, int j0, int m, int hi, v2f a,
                const float* __restrict__ nI, const float* __restrict__ rI) {
    const int j = j0 + m;                  // N = lane & 15
    const float pjx = pb[j * 3 + 0];
    const float pjy = pb[j * 3 + 1];
    const float pjz = pb[j * 3 + 2];
    const float rj  = rb[j];
    const float nj  = pjx * pjx + pjy * pjy + pjz * pjz;

    v2f bm;                                // B-matrix operand (4x16, f32)
    bm.x = hi ? pjz  : pjx;
    bm.y = hi ? 0.0f : pjy;

    v8f c = {};
    // D = A x B : 16x16 tile of dot(pi, pj)
    c = __builtin_amdgcn_wmma_f32_16x16x4_f32(
            /*neg_a=*/false, a, /*neg_b=*/false, bm,
            /*c_mod=*/(short)0, c, /*reuse_a=*/false, /*reuse_b=*/false);

    float acc = 0.0f;
    #pragma unroll
    for (int v = 0; v < 8; ++v) {
        // element (M, N): M = v + 8*hi, N = m
        float sq   = nI[v] + nj - 2.0f * c[v];
        float nrm  = __builtin_amdgcn_sqrtf(fmaxf(sq, 0.0f));
        float pr   = rI[v] + rj;
        float dist = nrm - pr;
        float dh   = ALPHA * pr;
        // log(max(dist,eps)/max(dh,eps)) = (log2(num) - log2(den)) * ln2
        float l2n  = __builtin_amdgcn_logf(fmaxf(dist, EPSILON));
        float l2d  = __builtin_amdgcn_logf(fmaxf(dh,   EPSILON));
        float lt   = (l2n - l2d) * LN2F;
        float dd   = dist - dh;
        float soft = -(dd * dd) * lt;
        float ed   = EPSILON - dist;
        float hard = ed * ed + CONSTANT;
        float bar  = (dist < dh)   ? soft : 0.0f;
        bar        = (dist < 0.0f) ? hard : bar;
        if (DIAG) {
            const int gi = i0 + v + 8 * hi;
            const int gj = j0 + m;
            bar = (gj > gi) ? bar : 0.0f;  // strict upper triangle on diagonal tile
        }
        acc += bar;
    }
    return acc;
}

// --- one row-tile of the collision barrier, per wave -------------------------
__device__ __forceinline__
float row_tiles(const float* __restrict__ pb,   // pos for batch b: [N][3]
                const float* __restrict__ rb,   // radius for batch b: [N]
                int i_tile, int wave, int lane) {
    const int m  = lane & 15;
    const int hi = lane >> 4;        // 0: lanes 0-15 (K=0,1), 1: lanes 16-31 (K=2,3)
    const int i0 = i_tile * 16;
    const int i  = i0 + m;

    // A-matrix operand (16x4, f32): VGPR0 = x|z, VGPR1 = y|0 per ISA layout.
    const float pix = pb[i * 3 + 0];
    const float piy = pb[i * 3 + 1];
    const float piz = pb[i * 3 + 2];
    const float ri  = rb[i];
    const float ni  = pix * pix + piy * piy + piz * piz;

    v2f a;
    a.x = hi ? piz  : pix;
    a.y = hi ? 0.0f : piy;

    // Broadcast ||pi||^2 and r_i to the 8 M-slots this lane owns (M = v + 8*hi).
    float nI[8], rI[8];
    #pragma unroll
    for (int v = 0; v < 8; ++v) {
        const int src = v + 8 * hi;        // lane `src` (<16) holds row m=src
        nI[v] = __shfl(ni, src, 32);
        rI[v] = __shfl(ri, src, 32);
    }

    float acc = 0.0f;
    // diagonal tile (needs j>i masking): wave 0 only — wave-uniform branch,
    // EXEC stays all-ones inside (WMMA requirement).
    if (wave == 0)
        acc += tile_eval<true>(pb, rb, i0, i0, m, hi, a, nI, rI);
    // off-diagonal tiles: no per-element masking.
    for (int j_tile = i_tile + 1 + wave; j_tile < TILES; j_tile += NWAVES)
        acc += tile_eval<false>(pb, rb, i0, j_tile * 16, m, hi, a, nI, rI);
    return acc;
}

// --- collision kernel: block = rows (t, T-1-t) of one batch ------------------
__global__ __launch_bounds__(TPB)
void collision_kernel(const float* __restrict__ pos,
                      const float* __restrict__ rad,
                      float* __restrict__ out) {
    const int half = TILES / 2;
    const int b    = blockIdx.x / half;
    const int t    = blockIdx.x % half;
    const int wave = threadIdx.x >> 5;
    const int lane = threadIdx.x & 31;

    const float* pb = pos + (size_t)b * NPTS * 3;
    const float* rb = rad + (size_t)b * NPTS;

    float acc = row_tiles(pb, rb, t, wave, lane);
    acc      += row_tiles(pb, rb, TILES - 1 - t, wave, lane);

    __shared__ float red[TPB];
    red[threadIdx.x] = acc;
    __syncthreads();
    #pragma unroll
    for (int s = TPB / 2; s > 0; s >>= 1) {
        if (threadIdx.x < s) red[threadIdx.x] += red[threadIdx.x + s];
        __syncthreads();
    }
    if (threadIdx.x == 0) {
        const float n_pairs = (float)BATCH * (float)NPTS * (float)(NPTS - 1) * 0.5f;
        atomicAdd(out, red[0] * (COLLISION_W / n_pairs));
    }
}

// ---------------------------------------------------------------------------
extern "C" void kernel_launch(void* const* d_in, const int* in_sizes, int n_in,
                              void* d_out, int out_size, void* d_ws, size_t ws_size,
                              hipStream_t stream) {
    (void)in_sizes; (void)n_in; (void)out_size; (void)d_ws; (void)ws_size;
    const float* pos = (const float*)d_in[0];
    const float* rad = (const float*)d_in[1];
    float* out = (float*)d_out;

    hipMemsetAsync(out, 0, sizeof(float), stream);
    grav_ground_kernel<<<(BATCH * NPTS) / 256, 256, 0, stream>>>(pos, rad, out);
    collision_kernel<<<BATCH * (TILES / 2), TPB, 0, stream>>>(pos, rad, out);
}